// TripletLoss_32925219291441
// MI455X (gfx1250) — compile-verified
//
#include <hip/hip_runtime.h>
#include <math.h>

typedef __attribute__((ext_vector_type(2))) float v2f;
typedef __attribute__((ext_vector_type(8))) float v8f;

#define WAVES_PER_BLOCK 8
#define TRIPS_PER_WAVE 16   // one 16x16 WMMA tile's diagonal per wave
#define D 128               // feature dim (reference: fixed at 128)

__global__ void tl_zero_acc(double* acc) { *acc = 0.0; }

__global__ __launch_bounds__(256) void tl_triplet_wmma(
    const float* __restrict__ X,     // [N,128] fp32
    const int*   __restrict__ trip,  // [T,3] int32
    double*      __restrict__ acc,
    int T, int ngroups)
{
    __shared__ float diagbuf[WAVES_PER_BLOCK][256];

    const int lane = threadIdx.x & 31;
    const int wave = threadIdx.x >> 5;
    const int group = blockIdx.x * WAVES_PER_BLOCK + wave;
    if (group >= ngroups) return;          // wave-uniform exit (EXEC stays full)

    const int m  = lane & 15;              // matrix row (A) / col (B) this lane serves
    const int hi = lane >> 4;              // half-wave: K offset +2 per ISA f32 layout
    int t = group * TRIPS_PER_WAVE + m;    // triplet handled by column m
    const int tc = (t < T) ? t : (T - 1);  // clamp tail reads

    const int ii = trip[3 * tc + 0];
    const int jj = trip[3 * tc + 1];
    const int kk = trip[3 * tc + 2];

    const float* rowI = X + (size_t)ii * D + 2 * hi;
    const float* rowJ = X + (size_t)jj * D + 2 * hi;
    const float* rowK = X + (size_t)kk * D + 2 * hi;

    v8f c1 = {};   // diag -> dot(x_i, x_j)
    v8f c2 = {};   // diag -> dot(x_i, x_k)
    float sqi = 0.f, sqj = 0.f, sqk = 0.f;

    // 128 = 32 chunks of K=4.  f32 WMMA A/B layout: lane(m,hi) holds
    // {elem[k0+2*hi], elem[k0+2*hi+1]} -> a single b64 load per matrix.
#pragma unroll
    for (int c = 0; c < 32; ++c) {
        v2f a  = *(const v2f*)(rowI + 4 * c);
        v2f bj = *(const v2f*)(rowJ + 4 * c);
        v2f bk = *(const v2f*)(rowK + 4 * c);
        sqi += a.x  * a.x  + a.y  * a.y;
        sqj += bj.x * bj.x + bj.y * bj.y;
        sqk += bk.x * bk.x + bk.y * bk.y;
        // D = A(16x4, f32) * B(4x16, f32) + C(16x16, f32)
        c1 = __builtin_amdgcn_wmma_f32_16x16x4_f32(
                 false, a, false, bj, (short)0, c1, false, false);
        c2 = __builtin_amdgcn_wmma_f32_16x16x4_f32(
                 false, a, false, bk, (short)0, c2, false, false);
    }

    // Each half-wave covered half the K range of its row -> combine.
    sqi += __shfl_xor(sqi, 16, 32);
    sqj += __shfl_xor(sqj, 16, 32);
    sqk += __shfl_xor(sqk, 16, 32);

    // Diagonal extraction. C/D layout: element (M,N) with M=N=t lives in
    // VGPR (t&7) at lane t (t<8) or lane t+16 (t>=8).  Stage through LDS
    // (same-wave DS ops are in-order -> wave-synchronous, no barrier).
    const int didx = ((m & 7) << 5) + ((m < 8) ? m : (m + 16));
#pragma unroll
    for (int v = 0; v < 8; ++v) diagbuf[wave][(v << 5) + lane] = c1[v];
    const float dot_ij = diagbuf[wave][didx];
#pragma unroll
    for (int v = 0; v < 8; ++v) diagbuf[wave][(v << 5) + lane] = c2[v];
    const float dot_ik = diagbuf[wave][didx];

    // dist = max(||i||^2 + ||x||^2 - 2<i,x>, 0)  (matches reference clamp)
    const float dij = fmaxf(sqi + sqj - 2.0f * dot_ij, 0.0f);
    const float dik = fmaxf(sqi + sqk - 2.0f * dot_ik, 0.0f);
    const float x   = dij - dik;
    // logaddexp(0,x) = max(x,0) + log1p(exp(-|x|))
    float sp = fmaxf(x, 0.0f) + log1pf(expf(-fabsf(x)));

    if (hi || t >= T) sp = 0.0f;   // hi half duplicates lo half; mask tail
    sp += __shfl_xor(sp, 1, 32);
    sp += __shfl_xor(sp, 2, 32);
    sp += __shfl_xor(sp, 4, 32);
    sp += __shfl_xor(sp, 8, 32);
    sp += __shfl_xor(sp, 16, 32);

    if (lane == 0) atomicAdd(acc, (double)sp);   // global_atomic_add_f64
}

__global__ void tl_finalize(const double* acc, float* out, double invT) {
    out[0] = (float)(*acc * invT);
}

extern "C" void kernel_launch(void* const* d_in, const int* in_sizes, int n_in,
                              void* d_out, int out_size, void* d_ws, size_t ws_size,
                              hipStream_t stream) {
    const float* X    = (const float*)d_in[0];   // [8192,128] fp32
    const int*   trip = (const int*)d_in[1];     // [T,3] int32
    const int T = in_sizes[1] / 3;

    double* acc = (double*)d_ws;
    tl_zero_acc<<<1, 1, 0, stream>>>(acc);

    const int ngroups = (T + TRIPS_PER_WAVE - 1) / TRIPS_PER_WAVE;
    const int blocks  = (ngroups + WAVES_PER_BLOCK - 1) / WAVES_PER_BLOCK;
    tl_triplet_wmma<<<blocks, WAVES_PER_BLOCK * 32, 0, stream>>>(
        X, trip, acc, T, ngroups);

    tl_finalize<<<1, 1, 0, stream>>>(acc, (float*)d_out, 1.0 / (double)T);
}